// KNNFeatureLayer_61675730370814
// MI455X (gfx1250) — compile-verified
//
#include <hip/hip_runtime.h>

#define B_   8
#define N_   1024
#define F_   64
#define TC   128            // candidates per LDS tile
#define TILES (N_ / TC)     // 8
#define BLK  256
#define QPB  256            // queries per block

// gfx1250 async global->LDS copy (ASYNCcnt-tracked). GV mode: vdst = LDS byte
// address (VGPR32), vaddr = 64-bit global address (VGPR pair), no saddr.
__device__ __forceinline__ void async_ld_b128(unsigned lds_off, const void* gaddr) {
  asm volatile("global_load_async_to_lds_b128 %0, %1, off"
               :: "v"(lds_off), "v"(gaddr)
               : "memory");
}
__device__ __forceinline__ void wait_async0() {
  asm volatile("s_wait_asynccnt 0" ::: "memory");
}

__global__ __launch_bounds__(BLK) void knn_l1_top2_kernel(
    const float* __restrict__ feat0,
    const float* __restrict__ feat1,
    const float* __restrict__ pos1,
    float* __restrict__ out)
{
  __shared__ float tile[2][TC * F_];   // 2 x 32 KB double buffer

  const int tid   = threadIdx.x;
  const int batch = blockIdx.x >> 2;          // 4 blocks per batch
  const int q     = ((blockIdx.x & 3) * QPB) + tid;

  // Query features -> 16 float4 registers (global_load_b128 x16)
  const float4* f0 = (const float4*)(feat0 + ((size_t)(batch * N_ + q)) * F_);
  float4 qr[16];
#pragma unroll
  for (int k = 0; k < 16; ++k) qr[k] = f0[k];

  const float* f1base = feat1 + (size_t)batch * N_ * F_;

  // Cooperative async stage of one 128x64 tile: 8 x b128 per thread.
  auto issue_tile = [&](int t, int buf) {
    const float4* src = (const float4*)(f1base + (size_t)t * TC * F_);
    float* dst = tile[buf];
#pragma unroll
    for (int r = 0; r < 8; ++r) {
      const int e = tid + BLK * r;  // float4 index within tile, 0..2047
      const unsigned lds = (unsigned)(unsigned long long)(dst + 4 * e);
      async_ld_b128(lds, (const void*)(src + e));
    }
  };

  float best0 = 3.4e38f, best1 = 3.4e38f;
  int   bidx  = 0;

  issue_tile(0, 0);
  wait_async0();
  __syncthreads();

  for (int t = 0; t < TILES; ++t) {
    const int buf = t & 1;
    if (t + 1 < TILES) issue_tile(t + 1, buf ^ 1);        // overlap copy w/ compute
    if (t + 2 < TILES)                                    // GL2 prefetch tile t+2
      __builtin_prefetch((const void*)(f1base + (size_t)(t + 2) * TC * F_ + tid * 8), 0, 0);

    const float* tb = tile[buf];
    for (int j = 0; j < TC; ++j) {
      // Wave-uniform LDS address -> broadcast reads, no bank conflicts.
      const float4* c = (const float4*)(tb + j * F_);
      float a0 = 0.f, a1 = 0.f, a2 = 0.f, a3 = 0.f;
#pragma unroll
      for (int k = 0; k < 16; k += 4) {
        const float4 c0 = c[k+0], c1 = c[k+1], c2 = c[k+2], c3 = c[k+3];
        a0 += __builtin_fabsf(qr[k+0].x - c0.x) + __builtin_fabsf(qr[k+0].y - c0.y)
            + __builtin_fabsf(qr[k+0].z - c0.z) + __builtin_fabsf(qr[k+0].w - c0.w);
        a1 += __builtin_fabsf(qr[k+1].x - c1.x) + __builtin_fabsf(qr[k+1].y - c1.y)
            + __builtin_fabsf(qr[k+1].z - c1.z) + __builtin_fabsf(qr[k+1].w - c1.w);
        a2 += __builtin_fabsf(qr[k+2].x - c2.x) + __builtin_fabsf(qr[k+2].y - c2.y)
            + __builtin_fabsf(qr[k+2].z - c2.z) + __builtin_fabsf(qr[k+2].w - c2.w);
        a3 += __builtin_fabsf(qr[k+3].x - c3.x) + __builtin_fabsf(qr[k+3].y - c3.y)
            + __builtin_fabsf(qr[k+3].z - c3.z) + __builtin_fabsf(qr[k+3].w - c3.w);
      }
      const float d  = (a0 + a1) + (a2 + a3);
      const int   jg = t * TC + j;

      // Branchless top-2, strict < matches top_k first-occurrence tie-break.
      const bool  lt0 = d < best0;
      const float nb1 = lt0 ? best0 : ((d < best1) ? d : best1);
      bidx  = lt0 ? jg : bidx;
      best0 = lt0 ? d  : best0;
      best1 = nb1;
    }

    if (t + 1 < TILES) { wait_async0(); __syncthreads(); }
  }

  // Outputs: matched_pos (B,N,2) then weights (B,N), concatenated flat.
  const int gq = batch * N_ + q;
  const float2 p = ((const float2*)pos1)[(size_t)batch * N_ + bidx];
  out[2 * gq + 0]     = p.x;
  out[2 * gq + 1]     = p.y;
  out[B_ * N_ * 2 + gq] = best0 / (best1 - best0);   // 1 / (d1/d0 - 1)
}

extern "C" void kernel_launch(void* const* d_in, const int* in_sizes, int n_in,
                              void* d_out, int out_size, void* d_ws, size_t ws_size,
                              hipStream_t stream) {
  const float* feat0 = (const float*)d_in[0];
  const float* feat1 = (const float*)d_in[1];
  const float* pos1  = (const float*)d_in[2];
  float* out = (float*)d_out;

  dim3 grid(B_ * (N_ / QPB));   // 32 blocks
  dim3 block(BLK);              // 256 threads = 8 wave32s
  hipLaunchKernelGGL(knn_l1_top2_kernel, grid, block, 0, stream,
                     feat0, feat1, pos1, out);
}